// CDLoss_8478265442710
// MI455X (gfx1250) — compile-verified
//
#include <hip/hip_runtime.h>
#include <hip/hip_bf16.h>

#define B_ 8
#define N_ 4096
#define NTILES (N_ / 16)      // 256
#define JT_HALF (NTILES / 2)  // 128 tiles per j-half

typedef __attribute__((ext_vector_type(2))) float v2f;
typedef __attribute__((ext_vector_type(8))) float v8f;

// Kernel 1: pred squared norms + init column-min keys to +inf bits.
// (Column minima are stored as raw IEEE bits of non-negative floats, which
//  compare correctly as u32 — no encode/decode needed.)
__global__ __launch_bounds__(256) void cd_init(const float* __restrict__ pred,
                                               float* __restrict__ pn,
                                               unsigned* __restrict__ colkey) {
    int idx = blockIdx.x * blockDim.x + threadIdx.x;   // exactly B_*N_ threads
    const float* p = pred + 3 * idx;
    pn[idx] = p[0]*p[0] + p[1]*p[1] + p[2]*p[2];
    colkey[idx] = 0x7F800000u;                          // +inf
}

// Kernel 2: wave = (batch, pred-tile-pair [32 rows], j-half [2048 gt pts]).
// gt j-half staged once into LDS in WMMA B-operand layout (+ gn computed in
// place); each B tile feeds TWO v_wmma_f32_16x16x4_f32. Col-mins: full
// distances (gn folded in, clamped >= 0) via one 16-lane u32 atomicMin per
// tile covering 32 i-rows. Row-mins: per-j-half partials to workspace.
__global__ __launch_bounds__(256) void cd_main(const float* __restrict__ pred,
                                               const float* __restrict__ pn,
                                               const float* __restrict__ gt,
                                               unsigned* __restrict__ colkey,
                                               float* __restrict__ rowp) {
    // BpS[jl*32 + l]      = (x, y) of local gt point jl*16+l   (lanes 0..15)
    // BpS[jl*32 + 16 + l] = (z, 0)                             (lanes 16..31)
    __shared__ v2f   BpS[JT_HALF * 32];   // 32 KB
    __shared__ float gnS[JT_HALF * 16];   //  8 KB

    const int lane  = threadIdx.x & 31;
    const int wave  = threadIdx.x >> 5;
    const int wi    = blockIdx.x * 8 + wave;    // 0..2047
    const int b     = wi >> 8;                  // batch      (uniform per block)
    const int r8    = wi & 255;
    const int jh    = r8 >> 7;                  // j-half     (uniform per block)
    const int ipair = r8 & 127;                 // pred tile pair
    const int lL    = lane & 15;
    const int half  = lane >> 4;

    const float* predb = pred + (size_t)b * N_ * 3;
    const float* gtb   = gt   + (size_t)b * N_ * 3;
    const float* pnb   = pn + b * N_;

    // ---- cooperative staging: 2048 gt points -> B panel + gn, in LDS ----
    const int jbase = jh * (JT_HALF * 16);
    for (int k = 0; k < 8; ++k) {
        int jj = (int)threadIdx.x + (k << 8);   // local point index 0..2047
        const float* gp = gtb + (size_t)(jbase + jj) * 3;
        float x = gp[0], y = gp[1], z = gp[2];
        int jt = jj >> 4, jl = jj & 15;
        v2f xy; xy.x = x; xy.y = y;
        v2f z0; z0.x = z; z0.y = 0.0f;
        BpS[jt * 32 + jl]      = xy;
        BpS[jt * 32 + 16 + jl] = z0;
        gnS[jj] = x*x + y*y + z*z;
    }
    __syncthreads();

    // A operands (16x4 f32: M = lane%16, K = (x,y,z,0) across {VGPR, half})
    const int it0 = ipair * 2, it1 = it0 + 1;
    const float* pp0 = predb + (size_t)(it0 * 16 + lL) * 3;
    const float* pp1 = predb + (size_t)(it1 * 16 + lL) * 3;
    v2f a0, a1;
    if (half == 0) { a0.x = pp0[0]; a0.y = pp0[1]; a1.x = pp1[0]; a1.y = pp1[1]; }
    else           { a0.x = pp0[2]; a0.y = 0.0f;   a1.x = pp1[2]; a1.y = 0.0f;   }

    // pred norms as aligned pairs (M = 2q(+1) + 8*half within each tile)
    const v2f* pnb0 = (const v2f*)(pnb + it0 * 16 + 8 * half);
    const v2f* pnb1 = (const v2f*)(pnb + it1 * 16 + 8 * half);
    v2f pnp0[4], pnp1[4];
#pragma unroll
    for (int q = 0; q < 4; ++q) { pnp0[q] = pnb0[q]; pnp1[q] = pnb1[q]; }

    float rm0[8], rm1[8];   // running min over this j-half of (gn_j - 2c)
#pragma unroll
    for (int r = 0; r < 8; ++r) { rm0[r] = __builtin_inff(); rm1[r] = __builtin_inff(); }

    unsigned* colkeyh = colkey + b * N_ + jh * (JT_HALF * 16);
    const v2f minus2 = { -2.0f, -2.0f };

    // rotated loop, start staggered by ipair; B-tile + gn software-pipelined
    int jl = ipair;
    v2f   bb  = BpS[jl * 32 + lane];
    float gnl = gnS[jl * 16 + lL];
    for (int t = 0; t < JT_HALF; ++t) {
        const int jln  = (jl + 1) & (JT_HALF - 1);
        v2f   bbn  = BpS[jln * 32 + lane];      // prefetch next B tile
        float gnln = gnS[jln * 16 + lL];

        v8f c0 = {}, c1 = {};
        c0 = __builtin_amdgcn_wmma_f32_16x16x4_f32(false, a0, false, bb,
                                                   (short)0, c0, false, false);
        c1 = __builtin_amdgcn_wmma_f32_16x16x4_f32(false, a1, false, bb,
                                                   (short)0, c1, false, false);

        v2f gn2; gn2.x = gnl; gn2.y = gnl;
        float cm = __builtin_inff();            // min over 32 M-rows of (pn - 2c)
#pragma unroll
        for (int q = 0; q < 4; ++q) {           // aligned pairs -> clean v_pk_fma
            v2f cc; cc.x = c0[2*q]; cc.y = c0[2*q + 1];
            v2f e1 = gn2     + cc * minus2;
            v2f e2 = pnp0[q] + cc * minus2;
            rm0[2*q]     = fminf(rm0[2*q],     e1.x);
            rm0[2*q + 1] = fminf(rm0[2*q + 1], e1.y);
            cm = fminf(fminf(cm, e2.x), e2.y);
        }
#pragma unroll
        for (int q = 0; q < 4; ++q) {
            v2f cc; cc.x = c1[2*q]; cc.y = c1[2*q + 1];
            v2f e1 = gn2     + cc * minus2;
            v2f e2 = pnp1[q] + cc * minus2;
            rm1[2*q]     = fminf(rm1[2*q],     e1.x);
            rm1[2*q + 1] = fminf(rm1[2*q + 1], e1.y);
            cm = fminf(fminf(cm, e2.x), e2.y);
        }
        cm = fminf(cm, __shfl_xor(cm, 16, 32));         // combine lane-halves
        float cmd = fmaxf(gnl + cm, 0.0f);              // true min distance >= 0
        if (half == 0)
            atomicMin(colkeyh + jl * 16 + lL, __float_as_uint(cmd));

        bb = bbn; gnl = gnln; jl = jln;
    }

    // Row-min partials: reduce across the 16 lanes of each half, then lane 0
    // of each half writes pn_i + min_j(gn_j - 2c) for its 16 rows.
#pragma unroll
    for (int off = 1; off < 16; off <<= 1) {
#pragma unroll
        for (int r = 0; r < 8; ++r) {
            rm0[r] = fminf(rm0[r], __shfl_xor(rm0[r], off, 32));
            rm1[r] = fminf(rm1[r], __shfl_xor(rm1[r], off, 32));
        }
    }
    if (lL == 0) {
        float* dst = rowp + (size_t)jh * B_ * N_ + b * N_;
#pragma unroll
        for (int q = 0; q < 4; ++q) {
            dst[it0 * 16 + 8 * half + 2*q]     = pnp0[q].x + rm0[2*q];
            dst[it0 * 16 + 8 * half + 2*q + 1] = pnp0[q].y + rm0[2*q + 1];
            dst[it1 * 16 + 8 * half + 2*q]     = pnp1[q].x + rm1[2*q];
            dst[it1 * 16 + 8 * half + 2*q + 1] = pnp1[q].y + rm1[2*q + 1];
        }
    }
}

// Kernel 3: deterministic final reduction -> scalar loss.
__global__ __launch_bounds__(256) void cd_final(const unsigned* __restrict__ colkey,
                                                const float* __restrict__ rowp,
                                                float* __restrict__ out) {
    __shared__ double sm[256];
    int t = threadIdx.x;
    double acc = 0.0;
    for (int i = t; i < B_ * N_; i += 256) {
        acc += (double)__uint_as_float(colkey[i]);           // col-min distances
        acc += (double)fminf(rowp[i], rowp[B_ * N_ + i]);    // row-min distances
    }
    sm[t] = acc;
    __syncthreads();
    for (int s = 128; s > 0; s >>= 1) {
        if (t < s) sm[t] += sm[t + s];
        __syncthreads();
    }
    if (t == 0) out[0] = (float)(sm[0] / (double)B_);
}

extern "C" void kernel_launch(void* const* d_in, const int* in_sizes, int n_in,
                              void* d_out, int out_size, void* d_ws, size_t ws_size,
                              hipStream_t stream) {
    const float* pred = (const float*)d_in[0];
    const float* gt   = (const float*)d_in[1];

    char* ws = (char*)d_ws;
    unsigned* colkey = (unsigned*)ws;                          // B*N u32
    float*    pn     = (float*)(ws + (size_t)B_ * N_ * 4);     // B*N f32
    float*    rowp   = (float*)(ws + (size_t)2 * B_ * N_ * 4); // 2 * B*N f32

    cd_init<<<(B_ * N_) / 256, 256, 0, stream>>>(pred, pn, colkey);
    cd_main<<<(B_ * 256) / 8, 256, 0, stream>>>(pred, pn, gt, colkey, rowp);
    cd_final<<<1, 256, 0, stream>>>(colkey, rowp, (float*)d_out);
}